// CoarseWarp_10453950398629
// MI455X (gfx1250) — compile-verified
//
#include <hip/hip_runtime.h>
#include <stdint.h>

#define BDIM 2
#define CDIM 64
#define HRR  254           // ref spatial
#define WRR  254
#define HOO  256           // output spatial
#define WOO  256
#define LTOT (HRR * WRR)   // 64516
#define NSTAGE (3 * WRR)   // 762 contiguous index entries per (b, oy)
#define CPB  (CDIM / 4)    // 16 channels per block (grid.z = 4)

// out[b,c,oy,ox] = sum over taps (i,j) of padded[b,c, i+ym, j+xm]
// with m = index_map[b, (oy-i)*WRR + (ox-j)], padded = reflect-pad-1 of ref.
__global__ __launch_bounds__(256)
void coarse_warp_kernel(const float* __restrict__ ref,
                        const int*   __restrict__ idxmap,
                        float*       __restrict__ out) {
    __shared__ int lds_idx[768];

    const int oy = blockIdx.x;     // 0..255
    const int b  = blockIdx.y;     // 0..1
    const int ox = threadIdx.x;    // 0..255

    // The 3 index rows needed (y = oy-2..oy) are contiguous in index_map.
    int l0 = (oy - 2) * WRR;
    if (l0 < 0) l0 = 0;
    if (l0 > LTOT - NSTAGE) l0 = LTOT - NSTAGE;

    // --- Async stage 762 ints global -> LDS (CDNA5 async path, ASYNCcnt) ---
    // GV form: 64-bit per-lane address in VGPR pair, saddr slot = "off".
    const uint64_t gbase   = (uint64_t)(uintptr_t)(idxmap + (size_t)b * LTOT + l0);
    const unsigned ldsbase = (unsigned)(uintptr_t)&lds_idx[0];   // LDS byte offset
#pragma unroll
    for (int it = 0; it < 3; ++it) {
        int slot = it * 256 + ox;
        if (slot > NSTAGE - 1) slot = NSTAGE - 1;       // duplicate tail loads: benign
        unsigned ldso = ldsbase + (unsigned)slot * 4u;  // LDS byte address (VDST)
        uint64_t ga   = gbase + (uint64_t)((unsigned)slot * 4u);
        asm volatile("global_load_async_to_lds_b32 %0, %1, off"
                     :: "v"(ldso), "v"(ga) : "memory");
    }
    asm volatile("s_wait_asynccnt 0" ::: "memory");
    __syncthreads();

    // --- Decode the 9 tap source offsets once; reuse across channels ---
    int   soff[9];
    float w[9];
#pragma unroll
    for (int i = 0; i < 3; ++i) {
        const int  y   = oy - i;
        const bool yok = (unsigned)y < (unsigned)HRR;
#pragma unroll
        for (int j = 0; j < 3; ++j) {
            const int  t  = i * 3 + j;
            const int  x  = ox - j;
            int   off = 0;
            float wt  = 0.0f;
            if (yok && ((unsigned)x < (unsigned)WRR)) {
                const int m  = lds_idx[y * WRR + x - l0];
                const int ym = m / WRR;
                const int xm = m - ym * WRR;
                int py = i + ym - 1;                    // padded -> ref coord
                int px = j + xm - 1;
                py = py < 0 ? -py : (py >= HRR ? 2 * HRR - 2 - py : py); // reflect
                px = px < 0 ? -px : (px >= WRR ? 2 * WRR - 2 - px : px);
                off = py * WRR + px;
                wt  = 1.0f;
            }
            soff[t] = off;
            w[t]    = wt;
        }
    }

    // --- Channel loop: 4 channels at a time -> 36 independent loads in flight ---
    const int    c0    = blockIdx.z * CPB;
    const float* rbase = ref + ((size_t)b * CDIM + c0) * (size_t)(HRR * WRR);
    float*       obase = out + (((size_t)b * CDIM + c0) * (size_t)HOO + oy) * (size_t)WOO + ox;

    for (int cc = 0; cc < CPB; cc += 4) {
        const float* rc0 = rbase + (size_t)(cc + 0) * (HRR * WRR);
        const float* rc1 = rbase + (size_t)(cc + 1) * (HRR * WRR);
        const float* rc2 = rbase + (size_t)(cc + 2) * (HRR * WRR);
        const float* rc3 = rbase + (size_t)(cc + 3) * (HRR * WRR);

        float v0[9], v1[9], v2[9], v3[9];
#pragma unroll
        for (int t = 0; t < 9; ++t) {                   // load phase: all independent
            v0[t] = rc0[soff[t]];
            v1[t] = rc1[soff[t]];
            v2[t] = rc2[soff[t]];
            v3[t] = rc3[soff[t]];
        }

        float s0 = 0.0f, s1 = 0.0f, s2 = 0.0f, s3 = 0.0f;
#pragma unroll
        for (int t = 0; t < 9; ++t) {                   // math phase
            s0 = fmaf(w[t], v0[t], s0);
            s1 = fmaf(w[t], v1[t], s1);
            s2 = fmaf(w[t], v2[t], s2);
            s3 = fmaf(w[t], v3[t], s3);
        }

        __builtin_nontemporal_store(s0, obase + (size_t)(cc + 0) * (HOO * WOO));
        __builtin_nontemporal_store(s1, obase + (size_t)(cc + 1) * (HOO * WOO));
        __builtin_nontemporal_store(s2, obase + (size_t)(cc + 2) * (HOO * WOO));
        __builtin_nontemporal_store(s3, obase + (size_t)(cc + 3) * (HOO * WOO));
    }
}

extern "C" void kernel_launch(void* const* d_in, const int* in_sizes, int n_in,
                              void* d_out, int out_size, void* d_ws, size_t ws_size,
                              hipStream_t stream) {
    (void)in_sizes; (void)n_in; (void)d_ws; (void)ws_size; (void)out_size;
    // d_in[0] = lr (unused, shape-only), d_in[1] = ref (f32), d_in[2] = index_map (i32)
    const float* ref    = (const float*)d_in[1];
    const int*   idxmap = (const int*)d_in[2];
    float*       out    = (float*)d_out;

    dim3 grid(HOO, BDIM, 4);   // 2048 blocks
    dim3 block(256);           // 8 wave32s
    coarse_warp_kernel<<<grid, block, 0, stream>>>(ref, idxmap, out);
}